// DynFilter_39668317946346
// MI455X (gfx1250) — compile-verified
//
#include <hip/hip_runtime.h>

namespace {
constexpr int kB   = 4;
constexpr int kFT  = 3;
constexpr int kH   = 256;
constexpr int kW   = 256;
constexpr int kK   = 75;            // k = fh*15 + fw*3 + ft
constexpr int kOD  = 16;
constexpr int kPixFloats = kK * kOD;   // 1200 floats = 4800 B per pixel
constexpr int kBufFloats = 1280;       // padded to 10 * 128 floats (5120 B)
constexpr int kWaves = 4;              // waves per block (128 threads)
constexpr int kNPix  = 4;              // pixels per wave (double-buffered)
constexpr int kPixels = kB * kH * kW;  // 262144
} // namespace

// 16-byte int vector type matching the builtin's expected pointee type.
typedef int v4i_ __attribute__((vector_size(16)));

// ---- CDNA5 async-copy primitives (ASYNCcnt-tracked) -----------------------
#if !defined(__AMDGCN__)
#define ASYNC_WAIT(imm) ((void)0)
#elif __has_builtin(__builtin_amdgcn_s_wait_asynccnt)
#define ASYNC_WAIT(imm) __builtin_amdgcn_s_wait_asynccnt(imm)
#else
#define ASYNC_WAIT(imm) asm volatile("s_wait_asynccnt %0" :: "i"(imm) : "memory")
#endif

// Issue the 4800 B weight block of one pixel as 10 x 512 B async copies
// (32 lanes x b128). Tail lanes (step 9, lane >= 12) are clamped to the
// pixel base address and land in the LDS pad region (floats 1200..1279).
__device__ __forceinline__ void issue_pixel_copy(const float* __restrict__ gpix,
                                                 float* lbuf, int lane) {
#if defined(__AMDGCN__)
#pragma unroll
  for (int s = 0; s < 10; ++s) {
    const int e = s * 128 + 4 * lane;                 // float index in block
    const float* g = gpix + (e < kPixFloats ? e : 0); // clamp tail (stay in-bounds)
    float* l = lbuf + e;
#if __has_builtin(__builtin_amdgcn_global_load_async_to_lds_b128)
    __builtin_amdgcn_global_load_async_to_lds_b128(
        (__attribute__((address_space(1))) v4i_*)(void*)g,   // global src (AS1)
        (__attribute__((address_space(3))) v4i_*)(void*)l,   // LDS dst (AS3)
        0, 0);
#else
    const unsigned ldsAddr = (unsigned)(unsigned long long)(void*)l; // LDS byte offset
    const unsigned long long gAddr = (unsigned long long)(const void*)g;
    asm volatile("global_load_async_to_lds_b128 %0, %1, off"
                 :: "v"(ldsAddr), "v"(gAddr)
                 : "memory");
#endif
  }
#else
  (void)gpix; (void)lbuf; (void)lane;
#endif
}

// ---------------------------------------------------------------------------
extern "C" __global__ void __launch_bounds__(kWaves * 32, 1)
dynfilter_stream_kernel(const float* __restrict__ x,
                        const float* __restrict__ f,
                        float* __restrict__ out) {
  __shared__ float lds[kWaves * 2 * kBufFloats];   // 40 KB: double buffer / wave

  const int tid  = threadIdx.x;
  const int wv   = tid >> 5;
  const int lane = tid & 31;
  const int kq   = lane >> 2;   // k residue class mod 8 owned by this lane
  const int c    = lane & 3;    // output quad: o = 4c .. 4c+3

  float* buf[2] = { &lds[(wv * 2 + 0) * kBufFloats],
                    &lds[(wv * 2 + 1) * kBufFloats] };

  const long long p0 = ((long long)blockIdx.x * kWaves + wv) * kNPix;

  // Prologue: start streaming pixel 0's weights.
  issue_pixel_copy(f + (size_t)p0 * kPixFloats, buf[0], lane);

#pragma unroll
  for (int i = 0; i < kNPix; ++i) {
    const long long p = p0 + i;

    // Prefetch next pixel's weights into the other buffer.
    if (i + 1 < kNPix)
      issue_pixel_copy(f + (size_t)(p + 1) * kPixFloats, buf[(i + 1) & 1], lane);

    const int w = (int)(p & (kW - 1));
    const int h = (int)((p >> 8) & (kH - 1));
    const int b = (int)(p >> 16);

    // Patch values for k = kq + 8s (x is L2/WGP$-resident; overlaps the copy).
    float patch[10];
#pragma unroll
    for (int s = 0; s < 10; ++s) {
      const int k = kq + 8 * s;
      float pv = 0.0f;
      if (k < kK) {
        const int fh = k / 15;
        const int r  = k - fh * 15;
        const int fw = r / 3;
        const int ft = r - fw * 3;
        const int yy = h + fh - 2;
        const int xx = w + fw - 2;
        if ((unsigned)yy < (unsigned)kH && (unsigned)xx < (unsigned)kW)
          pv = x[(((size_t)b * kFT + ft) * kH + yy) * kW + xx];
      }
      patch[s] = pv;
    }

    // Loads complete in order: <=10 outstanding means this pixel's 10 are done.
    if (i + 1 < kNPix) ASYNC_WAIT(10);
    else               ASYNC_WAIT(0);

    // Accumulate: each ds_load_b128 is f[k, 4c:4c+4] for k = kq + 8s.
    const float* fb = buf[i & 1];
    float ax = 0.f, ay = 0.f, az = 0.f, aw = 0.f;
#pragma unroll
    for (int s = 0; s < 10; ++s) {
      const int k = kq + 8 * s;
      if (k < kK) {
        const float4 fv = *(const float4*)(fb + s * 128 + 4 * lane);
        ax += patch[s] * fv.x;
        ay += patch[s] * fv.y;
        az += patch[s] * fv.z;
        aw += patch[s] * fv.w;
      }
    }

    // Reduce the 8 k-classes (lanes sharing c): xor masks 4, 8, 16 (wave32).
#pragma unroll
    for (int m = 4; m < 32; m <<= 1) {
      ax += __shfl_xor(ax, m, 32);
      ay += __shfl_xor(ay, m, 32);
      az += __shfl_xor(az, m, 32);
      aw += __shfl_xor(aw, m, 32);
    }

    if (lane < 4) {
      float4 o4; o4.x = ax; o4.y = ay; o4.z = az; o4.w = aw;
      *(float4*)(out + (size_t)p * kOD + 4 * c) = o4;
    }
  }
}

// ---------------------------------------------------------------------------
extern "C" void kernel_launch(void* const* d_in, const int* in_sizes, int n_in,
                              void* d_out, int out_size, void* d_ws, size_t ws_size,
                              hipStream_t stream) {
  (void)in_sizes; (void)n_in; (void)out_size; (void)d_ws; (void)ws_size;
  const float* x = (const float*)d_in[0];
  const float* f = (const float*)d_in[1];
  float* out = (float*)d_out;

  const int pixPerBlock = kWaves * kNPix;        // 16
  const int blocks = kPixels / pixPerBlock;      // 16384
  hipLaunchKernelGGL(dynfilter_stream_kernel, dim3(blocks), dim3(kWaves * 32),
                     0, stream, x, f, out);
}